// SinkhornLoRaRouter_2302102471510
// MI455X (gfx1250) — compile-verified
//
#include <hip/hip_runtime.h>
#include <hip/hip_bf16.h>
#include <math.h>

// SinkhornLoRaRouter for MI455X (gfx1250, wave32).
//
// Pipeline:
//   K1 grouped GEMM (fp32 WMMA 16x16x4) + exp -> cost (T x 16) in ws
//   K2 single-workgroup Sinkhorn -> converged d1 (16 floats) in ws
//   K3 per-token top-k over d1[l]*cost[t,l] + softmax/sigmoid scores
//
// Notes:
//  * d0[t] > 0 scales whole rows of the sinkhorn output -> does not change
//    per-row top-k ordering, so only d1 is materialized.
//  * softmax(logits) == cost / rowsum(cost) since cost = exp(logits).
//  * Assumes tokens_per_expert entries are multiples of 128 (true here: 1024),
//    so each 128-token block maps to a single expert.

typedef float v2f __attribute__((ext_vector_type(2)));
typedef float v8f __attribute__((ext_vector_type(8)));

#define H_DIM 2048
#define L_DIM 16
#define KC 512           // K-chunk staged in LDS
#define LDS_STRIDE 516   // 516 % 64 == 4 -> conflict-free b64 reads across 16 lanes
#define TOL_F 1e-4f
#define EPS_F 1e-8f
#define MAX_ITERS 10000

// ---------------------------------------------------------------------------
// Kernel 1: grouped GEMM (exact fp32 via V_WMMA_F32_16X16X4_F32) + exp.
// Block = 256 threads = 8 waves; each wave owns one 16x16 output tile.
// Block b covers tokens [b*128, b*128+128), all within one expert.
// ---------------------------------------------------------------------------
__global__ __launch_bounds__(256) void gemm_exp_kernel(
    const float* __restrict__ x, const float* __restrict__ w1,
    const int* __restrict__ tpe, float* __restrict__ cost, int T, int E)
{
    __shared__ float sB[L_DIM * LDS_STRIDE];   // w1[e] chunk, transposed [L][KC]

    const int tid = threadIdx.x;
    const int t0  = blockIdx.x * 128;

    // Expert for this block via prefix scan of tokens_per_expert.
    int e = 0, acc_t = 0;
    for (int i = 0; i < E; ++i) {
        int c = tpe[i];
        if (t0 >= acc_t && t0 < acc_t + c) e = i;
        acc_t += c;
    }
    const float* we = w1 + (size_t)e * H_DIM * L_DIM;

    const int lane = tid & 31;
    const int wave = tid >> 5;
    const int m    = lane & 15;    // row within tile (A) == col within tile (B)
    const int g    = lane >> 4;    // K pair selector: K = 2g + vgpr

    const int   row  = t0 + wave * 16 + m;
    const float* arow = x + (size_t)row * H_DIM + 2 * g;
    const float* brow = sB + m * LDS_STRIDE + 2 * g;

    v8f acc = {};
    for (int kc = 0; kc < H_DIM; kc += KC) {
        __syncthreads();
        // Stage w1[e][kc:kc+KC][:] transposed into LDS, coalesced global reads.
        for (int i = tid; i < KC * L_DIM; i += 256) {
            int h = i >> 4, l = i & 15;
            sB[l * LDS_STRIDE + h] = we[(size_t)(kc + h) * L_DIM + l];
        }
        __syncthreads();
        #pragma unroll 4
        for (int k = 0; k < KC; k += 4) {
            v2f a = *(const v2f*)(arow + kc + k);   // A: K = kc+k+2g, +1
            v2f b = *(const v2f*)(brow + k);        // B: same K pair, N = m
            acc = __builtin_amdgcn_wmma_f32_16x16x4_f32(
                false, a, false, b, (short)0, acc, false, false);
        }
    }

    // C layout: vgpr r -> M = r + 8g, N = m.  Store cost = exp(logits).
    const int ctile = t0 + wave * 16;
    #pragma unroll
    for (int r = 0; r < 8; ++r) {
        int orow = ctile + r + 8 * g;
        cost[(size_t)orow * L_DIM + m] = expf(acc[r]);
    }
}

// ---------------------------------------------------------------------------
// Kernel 2: Sinkhorn iteration on cost (T x 16); single workgroup (1024 thr),
// 8 rows/thread, shfl_xor wave reduction for the 16 column sums.
// Produces only the converged d1 (16 floats).
// ---------------------------------------------------------------------------
__global__ __launch_bounds__(1024) void sinkhorn_kernel(
    const float* __restrict__ cost, float* __restrict__ d1_out, int T)
{
    __shared__ float s_d1[L_DIM];
    __shared__ float s_part[32][L_DIM];
    __shared__ float s_diff[L_DIM];
    __shared__ int   s_done;

    const int tid  = threadIdx.x;
    const int lane = tid & 31;
    const int wave = tid >> 5;
    const int rpt  = T / 1024;                 // rows per thread

    if (tid < L_DIM) s_d1[tid] = 1.0f;
    if (tid == 0)    s_done = 0;

    const float invT = 1.0f / (float)T;
    const float invL = 1.0f / (float)L_DIM;

    for (int iter = 0; iter < MAX_ITERS; ++iter) {
        __syncthreads();                       // covers s_d1 writes
        float d1l[L_DIM];
        #pragma unroll
        for (int l = 0; l < L_DIM; ++l) d1l[l] = s_d1[l];

        float part[L_DIM];
        #pragma unroll
        for (int l = 0; l < L_DIM; ++l) part[l] = 0.0f;

        for (int r = 0; r < rpt; ++r) {
            const float* rowp = cost + (size_t)(tid * rpt + r) * L_DIM;
            float rowv[L_DIM];
            #pragma unroll
            for (int l = 0; l < L_DIM; ++l) rowv[l] = rowp[l];
            float dot = 0.0f;
            #pragma unroll
            for (int l = 0; l < L_DIM; ++l) dot = fmaf(rowv[l], d1l[l], dot);
            float d0n = invT / (dot + EPS_F);  // d0n for this row
            #pragma unroll
            for (int l = 0; l < L_DIM; ++l) part[l] = fmaf(rowv[l], d0n, part[l]);
        }

        // Reduce 16 partials within each wave, then across the 32 waves.
        #pragma unroll
        for (int l = 0; l < L_DIM; ++l) {
            float v = part[l];
            for (int off = 16; off > 0; off >>= 1) v += __shfl_xor(v, off, 32);
            if (lane == 0) s_part[wave][l] = v;
        }
        __syncthreads();

        if (tid < L_DIM) {
            float col = 0.0f;
            #pragma unroll
            for (int w = 0; w < 32; ++w) col += s_part[w][tid];
            float d1n = invL / (col + EPS_F);
            s_diff[tid] = fabsf(s_d1[tid] - d1n);
            s_d1[tid]   = d1n;
        }
        __syncthreads();
        if (tid == 0) {
            float err = 0.0f;
            #pragma unroll
            for (int l = 0; l < L_DIM; ++l) err += s_diff[l];
            s_done = (err * invL <= TOL_F) ? 1 : 0;
        }
        __syncthreads();
        if (s_done) break;
    }
    if (tid < L_DIM) d1_out[tid] = s_d1[tid];
}

// ---------------------------------------------------------------------------
// Kernel 3: per-token top-k over d1[l]*cost[t,l]; scores from softmax
// (cost/rowsum) or sigmoid (cost/(1+cost)) when k==1. k read on-device.
// Output: [T*k scores (float)] [T*k indices (int32)].
// ---------------------------------------------------------------------------
__global__ __launch_bounds__(256) void route_kernel(
    const float* __restrict__ cost, const float* __restrict__ d1,
    const int* __restrict__ topk_p, float* __restrict__ out, int T)
{
    __shared__ float s_d1[L_DIM];
    if (threadIdx.x < L_DIM) s_d1[threadIdx.x] = d1[threadIdx.x];
    __syncthreads();

    const int t = blockIdx.x * 256 + threadIdx.x;
    if (t >= T) return;
    const int k = *topk_p;

    float rowv[L_DIM], norm[L_DIM];
    float rowsum = 0.0f;
    const float* rowp = cost + (size_t)t * L_DIM;
    #pragma unroll
    for (int l = 0; l < L_DIM; ++l) { rowv[l] = rowp[l]; rowsum += rowv[l]; }
    #pragma unroll
    for (int l = 0; l < L_DIM; ++l) norm[l] = s_d1[l] * rowv[l];

    float* scores  = out;
    int*   idx_out = (int*)(out + (size_t)T * k);
    const float inv = 1.0f / rowsum;

    for (int kk = 0; kk < k; ++kk) {
        float best = -INFINITY; int bi = 0;
        #pragma unroll
        for (int l = 0; l < L_DIM; ++l)
            if (norm[l] > best) { best = norm[l]; bi = l; }   // first-max tiebreak
        norm[bi] = -INFINITY;
        float a = (k == 1) ? (rowv[bi] / (1.0f + rowv[bi]))   // sigmoid(logit)
                           : (rowv[bi] * inv);                 // softmax(logit)
        scores[(size_t)t * k + kk]  = a;
        idx_out[(size_t)t * k + kk] = bi;
    }
}

extern "C" void kernel_launch(void* const* d_in, const int* in_sizes, int n_in,
                              void* d_out, int out_size, void* d_ws, size_t ws_size,
                              hipStream_t stream)
{
    const float* x    = (const float*)d_in[0];
    const float* w1   = (const float*)d_in[1];
    const int*   tpe  = (const int*)d_in[2];
    const int*   topk = (const int*)d_in[3];

    const int T = in_sizes[0] / H_DIM;          // 8192
    const int E = in_sizes[2];                  // 8

    float* cost = (float*)d_ws;                 // T*16 floats (512 KB)
    float* d1   = cost + (size_t)T * L_DIM;     // 16 floats

    gemm_exp_kernel<<<T / 128, 256, 0, stream>>>(x, w1, tpe, cost, T, E);
    sinkhorn_kernel<<<1, 1024, 0, stream>>>(cost, d1, T);
    route_kernel<<<(T + 255) / 256, 256, 0, stream>>>(cost, d1, topk, (float*)d_out, T);
}